// PSAttention_46050639347857
// MI455X (gfx1250) — compile-verified
//
#include <hip/hip_runtime.h>
#include <hip/hip_bf16.h>
#include <stdint.h>

typedef float v2f __attribute__((ext_vector_type(2)));
typedef float v8f __attribute__((ext_vector_type(8)));

#define TOPK 8
#define PW 66                 // padded width/height (64 + 1 halo each side)
#define PPIX (PW * PW)        // 4356 padded pixels per batch image

// ---------------- kernel 0: zero-fill padded qT/kT (borders must be 0 every call)
__global__ __launch_bounds__(256) void k_fill(float* __restrict__ buf, int n) {
  int idx = blockIdx.x * 256 + threadIdx.x;
  if (idx < n) buf[idx] = 0.f;
}

// ---------------- kernel 1: transpose q,k -> padded pixel-major, channel-contiguous
__global__ __launch_bounds__(256) void k_transpose(const float* __restrict__ q,
                                                   const float* __restrict__ k,
                                                   float* __restrict__ qT,
                                                   float* __restrict__ kT) {
  int idx = blockIdx.x * 256 + threadIdx.x;   // over B*C*HW = 2*64*4096 = 524288
  int b = idx >> 18;
  int c = (idx >> 12) & 63;
  int p = idx & 4095;
  int y = p >> 6, x = p & 63;
  int o = ((size_t)0, (b * PPIX + (y + 1) * PW + (x + 1)) * 64 + c);
  qT[o] = q[idx];
  kT[o] = k[idx];
}

// ---------------- kernel 2: per-pixel channel sum-of-squares of kT (interior pixels)
__global__ __launch_bounds__(256) void k_sumsq(const float* __restrict__ kT,
                                               float* __restrict__ sk) {
  int bp = blockIdx.x * 256 + threadIdx.x;    // 0..8191
  int b = bp >> 12, p = bp & 4095;
  int y = p >> 6, x = p & 63;
  const float* r = kT + ((size_t)b * PPIX + (y + 1) * PW + (x + 1)) * 64;
  float s = 0.f;
#pragma unroll
  for (int i = 0; i < 64; i += 4) {
    float4 t = *(const float4*)(r + i);
    s += t.x * t.x + t.y * t.y + t.z * t.z + t.w * t.w;
  }
  sk[bp] = s;
}

// ---------------- kernel 3: zero-padded 3x3 box sum -> kn = ||kf_j||^2
__global__ __launch_bounds__(256) void k_boxsum(const float* __restrict__ sk,
                                                float* __restrict__ kn) {
  int bp = blockIdx.x * 256 + threadIdx.x;
  int b = bp >> 12, p = bp & 4095;
  int y = p >> 6, x = p & 63;
  float s = 0.f;
#pragma unroll
  for (int dy = -1; dy <= 1; ++dy)
#pragma unroll
    for (int dx = -1; dx <= 1; ++dx) {
      int yy = y + dy, xx = x + dx;
      if ((unsigned)yy < 64u && (unsigned)xx < 64u)
        s += sk[(b << 12) | (yy << 6) | xx];
    }
  kn[bp] = s;
}

// ---------------- main: WMMA distance GEMM + fused top-8 + softmax + gather
__global__ __launch_bounds__(256) void k_psattn(const float* __restrict__ qT,
                                                const float* __restrict__ kT,
                                                const float* __restrict__ knrm,
                                                const float* __restrict__ v,
                                                float* __restrict__ out) {
  __shared__ float tileBuf[8 * 256];   // per-wave 16x16 staging
  __shared__ float cs[16 * 128];       // candidate scores per row
  __shared__ int   ci[16 * 128];       // candidate indices per row
  __shared__ float lw[16 * TOPK];
  __shared__ int   lj[16 * TOPK];

  const int tid  = threadIdx.x;
  const int wave = tid >> 5;
  const int lane = tid & 31;
  const int half = lane >> 4;
  const int lr   = lane & 15;
  const int b    = blockIdx.y;
  const int m0   = blockIdx.x << 4;

  const float* qTb = qT + (size_t)b * PPIX * 64;
  const float* kTb = kT + (size_t)b * PPIX * 64;
  const float* knb = knrm + (((size_t)b) << 12);

  // A-side neighbor base pointers per 3x3 offset (padded: always in-bounds, borders = 0)
  const int rowp = m0 + lr;
  const int ry = rowp >> 6, rx = rowp & 63;
  const int coff = half << 1;   // K-split of lanes: upper half handles K+2,K+3
  const float* ap9[9];
#pragma unroll
  for (int o = 0; o < 9; ++o) {
    int dy = o / 3 - 1, dx = o % 3 - 1;
    ap9[o] = qTb + (size_t)((ry + dy + 1) * PW + (rx + dx + 1)) * 64 + coff;
  }

  float ts[TOPK];
  int   tj[TOPK];
#pragma unroll
  for (int t = 0; t < TOPK; ++t) { ts[t] = -3.0e38f; tj[t] = 0; }

  float* tb = tileBuf + wave * 256;

  for (int nt = wave; nt < 256; nt += 8) {
    const int n0 = nt << 4;
    const int colp = n0 + lr;
    const int cy = colp >> 6, cx = colp & 63;

    v8f acc = {0.f, 0.f, 0.f, 0.f, 0.f, 0.f, 0.f, 0.f};
#pragma unroll
    for (int o = 0; o < 9; ++o) {
      const int dy = o / 3 - 1, dx = o % 3 - 1;
      const float* ap = ap9[o];
      const float* bp = kTb + (size_t)((cy + dy + 1) * PW + (cx + dx + 1)) * 64 + coff;
#pragma unroll
      for (int ks = 0; ks < 16; ++ks) {
        v2f a  = *(const v2f*)(ap + ks * 4);
        v2f bb = *(const v2f*)(bp + ks * 4);
        acc = __builtin_amdgcn_wmma_f32_16x16x4_f32(
            false, a, false, bb, (short)0, acc, false, false);
      }
    }

    // stage accumulator tile to wave-private LDS (C layout: VGPR r -> row r + 8*half)
#pragma unroll
    for (int r = 0; r < 8; ++r)
      tb[((half << 3) + r) * 16 + lr] = acc[r];
    asm volatile("" ::: "memory");   // same-wave LDS is in-order; block compiler reorder

    // per-lane running top-8: lane owns row lr, columns [8*half, 8*half+8)
    const int cb = half << 3;
#pragma unroll
    for (int t = 0; t < 8; ++t) {
      int c = cb + t;
      float sc = 2.0f * tb[lr * 16 + c] - knb[n0 + c];   // = -cost + const(row)
      int jn = n0 + c;
      if (sc > ts[0]) {
        int pos = 0;
#pragma unroll
        for (int u = 1; u < TOPK; ++u)
          if (sc > ts[u]) { ts[u - 1] = ts[u]; tj[u - 1] = tj[u]; pos = u; }
        ts[pos] = sc; tj[pos] = jn;
      }
    }
    asm volatile("" ::: "memory");
  }

  // dump per-(wave,half) top-8 candidates
  {
    int base = lr * 128 + wave * 16 + (half << 3);
#pragma unroll
    for (int t = 0; t < TOPK; ++t) { cs[base + t] = ts[t]; ci[base + t] = tj[t]; }
  }
  __syncthreads();

  // per-row merge of 128 candidates + softmax
  if (tid < 16) {
    float fs[TOPK]; int fj[TOPK];
#pragma unroll
    for (int t = 0; t < TOPK; ++t) { fs[t] = -3.0e38f; fj[t] = 0; }
    const int base = tid * 128;
    for (int m = 0; m < 128; ++m) {
      float sc = cs[base + m]; int jn = ci[base + m];
      if (sc > fs[0]) {
        int pos = 0;
#pragma unroll
        for (int u = 1; u < TOPK; ++u)
          if (sc > fs[u]) { fs[u - 1] = fs[u]; fj[u - 1] = fj[u]; pos = u; }
        fs[pos] = sc; fj[pos] = jn;
      }
    }
    float mx = fs[TOPK - 1];                 // sorted ascending -> max at end
    float e[TOPK]; float sum = 0.f;
#pragma unroll
    for (int t = 0; t < TOPK; ++t) { e[t] = __expf(fs[t] - mx); sum += e[t]; }
    float inv = 1.0f / sum;
#pragma unroll
    for (int t = 0; t < TOPK; ++t) {
      lw[tid * TOPK + t] = e[t] * inv;
      lj[tid * TOPK + t] = fj[t];
    }
  }
  __syncthreads();

  // weighted gather of v: thread -> (row, 4 channels)
  {
    int row = tid >> 4;
    int c0  = (tid & 15) << 2;
    int pix = m0 + row;
    const float* vb = v + (((size_t)b * 64 + c0) << 12);
    float a0 = 0.f, a1 = 0.f, a2 = 0.f, a3 = 0.f;
#pragma unroll
    for (int t = 0; t < TOPK; ++t) {
      float w = lw[row * TOPK + t];
      int   j = lj[row * TOPK + t];
      a0 += w * vb[j];
      a1 += w * vb[(1 << 12) + j];
      a2 += w * vb[(2 << 12) + j];
      a3 += w * vb[(3 << 12) + j];
    }
    float* ob = out + (((size_t)b * 64 + c0) << 12) + pix;
    ob[0] = a0; ob[1 << 12] = a1; ob[2 << 12] = a2; ob[3 << 12] = a3;
  }
}

extern "C" void kernel_launch(void* const* d_in, const int* in_sizes, int n_in,
                              void* d_out, int out_size, void* d_ws, size_t ws_size,
                              hipStream_t stream) {
  (void)in_sizes; (void)n_in; (void)out_size; (void)ws_size;
  const float* q = (const float*)d_in[0];
  const float* k = (const float*)d_in[1];
  const float* v = (const float*)d_in[2];
  float* out = (float*)d_out;

  const int padElems = 2 * PPIX * 64;       // 557568 floats per array
  float* ws = (float*)d_ws;
  float* qT = ws;                           // padded
  float* kT = qT + padElems;                // padded
  float* sk = kT + padElems;                // 2*4096
  float* kn = sk + 2 * 4096;                // 2*4096   (~4.6 MB total)

  int fillN = 2 * padElems;                 // zero both padded arrays every call
  k_fill<<<(fillN + 255) / 256, 256, 0, stream>>>(qT, fillN);
  k_transpose<<<2048, 256, 0, stream>>>(q, k, qT, kT);
  k_sumsq<<<32, 256, 0, stream>>>(kT, sk);
  k_boxsum<<<32, 256, 0, stream>>>(sk, kn);
  dim3 grid(256, 2);
  k_psattn<<<grid, 256, 0, stream>>>(qT, kT, kn, v, out);
}